// TiedGraphAutoencoder_32427003085613
// MI455X (gfx1250) — compile-verified
//
#include <hip/hip_runtime.h>

// ---------------------------------------------------------------------------
// TiedGraphAutoencoder fused kernel for gfx1250 (MI455X, CDNA5, wave32).
//   out[b,m,c] = relu(Xsum . W_inner[deg] + b_inner[deg])   (0 if deg==5)
//              + relu(Xself . W_self     + b_self)
// computed with V_WMMA_F32_16X16X4_F32 (full fp32, K=68 = 17 x 4 steps).
// One block per batch element; 8 wave32s; wave w owns rows [16w,16w+16).
// Weights (6 x 68 x 128 f32 = 204KB) + atoms[b] (31.7KB) staged in LDS
// (320KB/WGP on CDNA5).
// ---------------------------------------------------------------------------

typedef __attribute__((ext_vector_type(2))) float v2f;
typedef __attribute__((ext_vector_type(8))) float v8f;

#define BDIM   128   // M: nodes per batch element
#define NF_A   62    // atom features
#define NF_B   6     // bond features
#define NDEG   5     // D: max degree slots
#define FAN    68    // NF_A + NF_B
#define NC     128   // output channels
#define KSTEPS 17    // FAN / 4

// LDS partition (float offsets)
#define SW_OFF     0                        // 6*68*128 = 52224 floats
#define SBIAS_OFF  (6 * FAN * NC)           // 52224
#define SATOMS_OFF (SBIAS_OFF + 6 * NC)     // 52992
#define SMEM_FLOATS (SATOMS_OFF + BDIM * NF_A)  // 60928 floats = 243712 B

__global__ __launch_bounds__(256) void tied_graph_ae_kernel(
    const float* __restrict__ atoms,    // (B, 128, 62)
    const float* __restrict__ bonds,    // (B, 128, 5, 6)
    const int*   __restrict__ edges,    // (B, 128, 5)  in [-1, 127]
    const float* __restrict__ W_inner,  // (5, 68, 128)
    const float* __restrict__ b_inner,  // (5, 128)
    const float* __restrict__ W_self,   // (68, 128)
    const float* __restrict__ b_self,   // (128)
    float*       __restrict__ out)      // (B, 128, 128)
{
    extern __shared__ float smem[];
    float* sW     = smem + SW_OFF;      // d in [0,4]: W_inner[d]; d==5: W_self
    float* sBias  = smem + SBIAS_OFF;   // d in [0,4]: b_inner[d]; d==5: b_self
    float* sAtoms = smem + SATOMS_OFF;  // atoms for this batch element

    const int b   = blockIdx.x;
    const int tid = threadIdx.x;

    // ---- cooperative staging into LDS -------------------------------------
    for (int i = tid; i < NDEG * FAN * NC; i += 256) sW[i] = W_inner[i];
    for (int i = tid; i < FAN * NC;        i += 256) sW[NDEG * FAN * NC + i] = W_self[i];
    for (int i = tid; i < NDEG * NC;       i += 256) sBias[i] = b_inner[i];
    for (int i = tid; i < NC;              i += 256) sBias[NDEG * NC + i] = b_self[i];
    for (int i = tid; i < BDIM * NF_A;     i += 256)
        sAtoms[i] = atoms[(long)b * BDIM * NF_A + i];
    __syncthreads();

    const int wave = tid >> 5;
    const int lane = tid & 31;
    const int hi   = lane >> 4;   // which K/M half this lane covers
    const int l15  = lane & 15;
    const int r0   = wave * 16;   // first row of this wave's M-tile
    const int row  = r0 + l15;    // A-matrix row owned by this lane (0..127)

    // ---- per-row edges / degree ------------------------------------------
    const long ebase = ((long)b * BDIM + row) * NDEG;
    int e[NDEG];
    int deg = 0;
#pragma unroll
    for (int d = 0; d < NDEG; ++d) {
        e[d] = edges[ebase + d];
        deg += (e[d] >= 0) ? 1 : 0;
    }

    // degree of each of the 8 output rows this lane holds in D-fragments
    int degv[8];
#pragma unroll
    for (int v = 0; v < 8; ++v)
        degv[v] = __shfl(deg, v + 8 * hi, 32);

    // ---- build A fragments in registers ----------------------------------
    // fp32 16x4 A layout: lane l<16 holds A[l][k0],A[l][k0+1];
    // lane l+16 holds A[l][k0+2],A[l][k0+3].  17 K-steps cover f = 0..67.
    float aS[2 * KSTEPS];   // summed neighbor features (Xsum)
    float aX[2 * KSTEPS];   // self features            (Xself)
    const long bbase = ((long)b * BDIM + row) * NDEG * NF_B;
#pragma unroll
    for (int t = 0; t < 2 * KSTEPS; ++t) {
        const int s = t >> 1;
        const int j = t & 1;
        const int f = 4 * s + 2 * hi + j;      // feature index 0..67
        float xs, xf;
        if (f < NF_A) {
            float acc = 0.f;
#pragma unroll
            for (int d = 0; d < NDEG; ++d)
                if (e[d] >= 0) acc += sAtoms[e[d] * NF_A + f];
            xs = acc;
            xf = sAtoms[row * NF_A + f];
        } else {
            const int fb = f - NF_A;
            float acc = 0.f;
#pragma unroll
            for (int d = 0; d < NDEG; ++d)
                acc += bonds[bbase + d * NF_B + fb];
            xs = acc;   // summed bonds feed both Xsum and Xself
            xf = acc;
        }
        aS[t] = xs;
        aX[t] = xf;
    }

    // ---- GEMM: 6 weight matrices x 8 column tiles of 16 -------------------
    for (int nt = 0; nt < NC / 16; ++nt) {
        const int n0 = nt * 16;
        float fin[8];
#pragma unroll
        for (int v = 0; v < 8; ++v) fin[v] = 0.f;

#pragma unroll
        for (int d = 0; d < 6; ++d) {
            const float* Wd = sW + d * FAN * NC;
            const float  bc = sBias[d * NC + n0 + l15];
            v8f c;
#pragma unroll
            for (int v = 0; v < 8; ++v) c[v] = bc;   // bias broadcast per column

#pragma unroll
            for (int s = 0; s < KSTEPS; ++s) {
                const int kr = 4 * s + 2 * hi;       // B 4x16: lane half picks K rows
                v2f bf;
                bf.x = Wd[kr * NC + n0 + l15];
                bf.y = Wd[(kr + 1) * NC + n0 + l15];
                v2f af;
                if (d < 5) { af.x = aS[2 * s]; af.y = aS[2 * s + 1]; }
                else       { af.x = aX[2 * s]; af.y = aX[2 * s + 1]; }
                // D = A(16x4,f32) * B(4x16,f32) + C ; full fp32 WMMA
                c = __builtin_amdgcn_wmma_f32_16x16x4_f32(
                        false, af, false, bf, (short)0, c, false, false);
            }

            if (d < 5) {
                // inner path: select rows whose degree == d, then relu
#pragma unroll
                for (int v = 0; v < 8; ++v)
                    fin[v] += (degv[v] == d) ? fmaxf(c[v], 0.f) : 0.f;
            } else {
                // self path: always added
#pragma unroll
                for (int v = 0; v < 8; ++v)
                    fin[v] += fmaxf(c[v], 0.f);
            }
        }

        // D-fragment: VGPR v holds row (v + 8*hi), column n0 + l15 — coalesced
#pragma unroll
        for (int v = 0; v < 8; ++v) {
            const int orow = r0 + v + 8 * hi;
            out[((long)b * BDIM + orow) * NC + n0 + l15] = fin[v];
        }
    }
}

extern "C" void kernel_launch(void* const* d_in, const int* in_sizes, int n_in,
                              void* d_out, int out_size, void* d_ws, size_t ws_size,
                              hipStream_t stream) {
    const float* atoms   = (const float*)d_in[0];
    const float* bonds   = (const float*)d_in[1];
    const int*   edges   = (const int*)d_in[2];
    const float* W_inner = (const float*)d_in[3];
    const float* b_inner = (const float*)d_in[4];
    const float* W_self  = (const float*)d_in[5];
    const float* b_self  = (const float*)d_in[6];
    float*       out     = (float*)d_out;

    const int B = in_sizes[0] / (BDIM * NF_A);          // 2048
    const size_t shmem = SMEM_FLOATS * sizeof(float);   // ~238 KB (CDNA5: 320KB/WGP)

    // allow >64KB dynamic LDS on this kernel (host-side attribute, capture-safe)
    hipFuncSetAttribute((const void*)tied_graph_ae_kernel,
                        hipFuncAttributeMaxDynamicSharedMemorySize, (int)shmem);

    tied_graph_ae_kernel<<<B, 256, shmem, stream>>>(
        atoms, bonds, edges, W_inner, b_inner, W_self, b_self, out);
}